// Kernel_Conv_83554293776951
// MI455X (gfx1250) — compile-verified
//
#include <hip/hip_runtime.h>

typedef __attribute__((ext_vector_type(2))) float v2f;
typedef __attribute__((ext_vector_type(8))) float v8f;

#define IN_CH   256
#define OUT_CH  512
#define BATCH   32
#define HW      16
#define KTOT    (IN_CH * IN_CH)   // 65536 (k = c*256 + m)
#define NK      32                // K-split chunks
#define KC      (KTOT / NK)       // 2048 k per block
#define KW      (KC / 8)          // 256 k per wave

// ---------------------------------------------------------------------------
// Phase 1: per-batch, compute
//   xs[b][c]        = sum_hw x[b,c,hw]
//   y[m][hw]        = relu(b1[m] + sum_c x[b,c,hw]*W1[m,c])      (LDS only)
//   S[b][c*256+m]   = sum_hw x[b,c,hw]*y[m,hw]
// ---------------------------------------------------------------------------
__global__ __launch_bounds__(256) void phase1_build_S(
    const float* __restrict__ x, const float* __restrict__ W1,
    const float* __restrict__ b1, float* __restrict__ S,
    float* __restrict__ xs)
{
    __shared__ float xl[IN_CH][HW];   // 16 KB
    __shared__ float yl[IN_CH][HW];   // 16 KB
    const int b = blockIdx.x;
    const int t = threadIdx.x;

    const float* xb = x + (size_t)b * IN_CH * HW;
    for (int i = t; i < IN_CH * HW; i += 256)
        xl[i >> 4][i & 15] = xb[i];
    __syncthreads();

    // spatial sums
    {
        float s = 0.f;
        #pragma unroll
        for (int hw = 0; hw < HW; ++hw) s += xl[t][hw];
        xs[b * IN_CH + t] = s;
    }

    // y row m = t
    {
        const int m = t;
        float acc[HW];
        const float bias = b1[m];
        #pragma unroll
        for (int hw = 0; hw < HW; ++hw) acc[hw] = bias;
        const float* w1r = W1 + (size_t)m * IN_CH;
        for (int c = 0; c < IN_CH; ++c) {
            const float w = w1r[c];
            #pragma unroll
            for (int hw = 0; hw < HW; ++hw) acc[hw] += xl[c][hw] * w;
        }
        #pragma unroll
        for (int hw = 0; hw < HW; ++hw) yl[m][hw] = fmaxf(acc[hw], 0.f);
    }
    __syncthreads();

    // S[b][c*256 + m], m = t  (coalesced stores across t)
    {
        const int m = t;
        float ym[HW];
        #pragma unroll
        for (int hw = 0; hw < HW; ++hw) ym[hw] = yl[m][hw];
        float* Sb = S + (size_t)b * KTOT;
        for (int c = 0; c < IN_CH; ++c) {
            float s = 0.f;
            #pragma unroll
            for (int hw = 0; hw < HW; ++hw) s += xl[c][hw] * ym[hw];
            Sb[c * IN_CH + m] = s;
        }
    }
}

// ---------------------------------------------------------------------------
// Phase 2: C[32 x 512] = S[32 x 65536] * W2^T, W2 viewed as [512 x 65536]
// row-major (W2_flat[o*65536 + k]).  K split across gridDim.y; 8 waves per
// block each cover KW=256 k with V_WMMA_F32_16X16X4_F32; LDS cross-wave
// reduction; deterministic partial writes (no atomics).
//
// Fragment layouts (ISA 7.12.2, 32-bit):
//   A 16x4 : lanes 0-15 -> M=lane, {v0=K0, v1=K1}; lanes 16-31 -> {K2, K3}
//   B 4x16 : lanes 0-15 -> N=lane, {v0=K0, v1=K1}; lanes 16-31 -> {K2, K3}
//   C 16x16: VGPR r: lanes 0-15 -> M=r, N=lane; lanes 16-31 -> M=r+8
// ---------------------------------------------------------------------------
__global__ __launch_bounds__(256) void phase2_wmma_gemm(
    const float* __restrict__ S, const float* __restrict__ W2,
    float* __restrict__ partials)
{
    const int ntile  = blockIdx.x;          // 0..31  (o tile of 16)
    const int kchunk = blockIdx.y;          // 0..NK-1
    const int lane   = threadIdx.x & 31;
    const int wave   = threadIdx.x >> 5;    // 0..7
    const int mrow   = lane & 15;           // M (A) / N (B) index
    const int koff   = (lane >> 4) * 2;     // K sub-offset within 4-wide step
    const int obase  = ntile * 16;

    const int kbeg = kchunk * KC + wave * KW;
    const int kend = kbeg + KW;

    const float* Sa0 = S  + (size_t)mrow * KTOT;            // b = mrow
    const float* Sa1 = S  + (size_t)(mrow + 16) * KTOT;     // b = mrow+16
    const float* Bw  = W2 + (size_t)(obase + mrow) * KTOT;  // o = obase+mrow

    v8f acc0 = {};   // C tile for b = 0..15
    v8f acc1 = {};   // C tile for b = 16..31

    for (int k0 = kbeg; k0 < kend; k0 += 16) {
        // stream W2 (the HBM-bound operand) ahead by 2 KB
        __builtin_prefetch(Bw + k0 + 512, 0, 0);
        #pragma unroll
        for (int kk = 0; kk < 16; kk += 4) {
            const int k = k0 + kk + koff;
            v2f a0 = *(const v2f*)(Sa0 + k);
            v2f a1 = *(const v2f*)(Sa1 + k);
            v2f bb = *(const v2f*)(Bw  + k);
            acc0 = __builtin_amdgcn_wmma_f32_16x16x4_f32(
                       false, a0, false, bb, (short)0, acc0, false, false);
            acc1 = __builtin_amdgcn_wmma_f32_16x16x4_f32(
                       false, a1, false, bb, (short)0, acc1, false, false);
        }
    }

    // cross-wave reduction in LDS (16 KB)
    __shared__ float red[8][512];
    #pragma unroll
    for (int r = 0; r < 8; ++r) {
        red[wave][r * 32 + lane]       = acc0[r];
        red[wave][256 + r * 32 + lane] = acc1[r];
    }
    __syncthreads();

    for (int idx = threadIdx.x; idx < 512; idx += 256) {
        float s = 0.f;
        #pragma unroll
        for (int w = 0; w < 8; ++w) s += red[w][idx];
        const int tile = idx >> 8;
        const int rr   = (idx >> 5) & 7;
        const int ln   = idx & 31;
        const int b    = tile * 16 + rr + ((ln >= 16) ? 8 : 0);
        const int ol   = ln & 15;
        partials[(((size_t)kchunk * 32 + ntile) * 32 + b) * 16 + ol] = s;
    }
}

// ---------------------------------------------------------------------------
// Phase 3: out[b,o] = sum_kc partials + sum_c xs[b,c]*b2[o*256+c]
// ---------------------------------------------------------------------------
__global__ __launch_bounds__(256) void phase3_reduce_bias(
    const float* __restrict__ partials, const float* __restrict__ xs,
    const float* __restrict__ b2, float* __restrict__ out)
{
    const int tid = blockIdx.x * 256 + threadIdx.x;  // 0..16383
    const int b = tid >> 9;
    const int o = tid & 511;

    const float* xsb = xs + b * IN_CH;
    const float* b2r = b2 + (size_t)o * IN_CH;
    float bias = 0.f;
    for (int c = 0; c < IN_CH; ++c) bias += xsb[c] * b2r[c];

    const int ntile = o >> 4, ol = o & 15;
    float s = 0.f;
    #pragma unroll 4
    for (int kc = 0; kc < NK; ++kc)
        s += partials[(((size_t)kc * 32 + ntile) * 32 + b) * 16 + ol];

    out[tid] = bias + s;
}

// ---------------------------------------------------------------------------
extern "C" void kernel_launch(void* const* d_in, const int* in_sizes, int n_in,
                              void* d_out, int out_size, void* d_ws, size_t ws_size,
                              hipStream_t stream) {
    const float* x  = (const float*)d_in[0];
    const float* W1 = (const float*)d_in[1];
    const float* b1 = (const float*)d_in[2];
    const float* W2 = (const float*)d_in[3];
    const float* b2 = (const float*)d_in[4];
    // d_in[5] (Wa), d_in[6] (ba): attention branch is dead code in the reference.
    float* out = (float*)d_out;

    float* ws       = (float*)d_ws;
    float* S        = ws;                                  // 32*65536 floats (8 MB)
    float* xs       = S  + (size_t)BATCH * KTOT;           // 8192 floats
    float* partials = xs + (size_t)BATCH * IN_CH;          // NK*32*32*16 = 524288 floats (2 MB)

    phase1_build_S<<<BATCH, 256, 0, stream>>>(x, W1, b1, S, xs);

    dim3 g2(OUT_CH / 16, NK);   // 32 x 32 blocks
    phase2_wmma_gemm<<<g2, 256, 0, stream>>>(S, W2, partials);

    phase3_reduce_bias<<<(BATCH * OUT_CH) / 256, 256, 0, stream>>>(partials, xs, b2, out);
}